// Transducer_6176162971720
// MI455X (gfx1250) — compile-verified
//
#include <hip/hip_runtime.h>
#include <hip/hip_bf16.h>
#include <math.h>

// Problem constants (from reference): B,T,U,H,V
#define B_ 4
#define T_ 256
#define U_ 48
#define H_ 512
#define V_ 1024
#define BLANK_ (V_ - 1)

typedef __attribute__((ext_vector_type(2))) float v2f;
typedef __attribute__((ext_vector_type(4))) float v4f;
typedef __attribute__((ext_vector_type(8))) float v8f;

// ---------------------------------------------------------------------------
// Kernel 1: P[m, v] = sum_h X[m,h] * W[v,h] (+ bias[v])
// X: (M, H) row-major, W: (V, H) row-major -> NT GEMM via V_WMMA_F32_16X16X4_F32
// Each wave32 computes a 16x32 output tile (2 N-tiles, shared A load).
// Per ISA layouts, both the 16x4 A tile and 4x16 B tile are a contiguous
// float2 per lane at row*H + k + 2*(lane>>4).
// ---------------------------------------------------------------------------
#define GEMM_WAVES 4
#define GEMM_NT 2   // N tiles per wave (A reuse)

__global__ __launch_bounds__(32 * GEMM_WAVES)
void rnnt_wmma_gemm_nt(const float* __restrict__ X, const float* __restrict__ W,
                       const float* __restrict__ bias, float* __restrict__ P,
                       int M) {
  const int lane  = threadIdx.x & 31;
  const int wave  = threadIdx.x >> 5;
  const int mbase = blockIdx.x * 16;
  const int nbase = (blockIdx.y * GEMM_WAVES + wave) * (16 * GEMM_NT);
  const int rloc  = lane & 15;
  const int khalf = (lane >> 4) << 1;   // lanes 16..31 hold K+2,K+3

  const bool fullM = (mbase + 16 <= M); // uniform: fast path for full tiles
  int mrow = mbase + rloc;
  if (!fullM && mrow >= M) mrow = M - 1;

  const float* xp  = X + (size_t)mrow * H_ + khalf;
  const float* wp0 = W + (size_t)(nbase + rloc) * H_ + khalf;
  const float* wp1 = wp0 + (size_t)16 * H_;

  v8f acc0 = {}, acc1 = {};
#pragma unroll 4
  for (int k = 0; k < H_; k += 4) {
    v2f a  = *reinterpret_cast<const v2f*>(xp + k);
    v2f b0 = *reinterpret_cast<const v2f*>(wp0 + k);
    v2f b1 = *reinterpret_cast<const v2f*>(wp1 + k);
    // D = A(16x4,f32) x B(4x16,f32) + C(16x16,f32)
    acc0 = __builtin_amdgcn_wmma_f32_16x16x4_f32(
        false, a, false, b0, (short)0, acc0, false, false);
    acc1 = __builtin_amdgcn_wmma_f32_16x16x4_f32(
        false, a, false, b1, (short)0, acc1, false, false);
  }

  const float bv0 = bias ? bias[nbase + rloc] : 0.0f;
  const float bv1 = bias ? bias[nbase + 16 + rloc] : 0.0f;
  const int mofs  = (lane >> 4) << 3;   // upper half-lanes own rows +8
  float* p0 = P + (size_t)(mbase + mofs) * V_ + nbase + rloc;

  if (fullM) {
#pragma unroll
    for (int r = 0; r < 8; ++r) {
      p0[(size_t)r * V_]      = acc0[r] + bv0;
      p0[(size_t)r * V_ + 16] = acc1[r] + bv1;
    }
  } else {
#pragma unroll
    for (int r = 0; r < 8; ++r) {
      if (mbase + mofs + r < M) {
        p0[(size_t)r * V_]      = acc0[r] + bv0;
        p0[(size_t)r * V_ + 16] = acc1[r] + bv1;
      }
    }
  }
}

// ---------------------------------------------------------------------------
// Kernel 2: per (b,u) block: stream over t, compute
//   lse(b,t,u) = logsumexp_v( LE[b,t,v] + LD[b,u,v] )
//   lp_blank(b,t,u) = x[BLANK] - lse ;  lp_label(b,t,u) = x[targets[b,u]] - lse
// LD row (4KB) pinned in registers (float4/thread); LE rows stream from L2.
// ---------------------------------------------------------------------------
__global__ __launch_bounds__(256)
void rnnt_lse(const float* __restrict__ LE, const float* __restrict__ LD,
              const int* __restrict__ targets,
              float* __restrict__ lpb, float* __restrict__ lpl) {
  const int bu  = blockIdx.x;
  const int b   = bu / (U_ + 1);
  const int u   = bu % (U_ + 1);
  const int tid = threadIdx.x;
  const int lane = tid & 31, wave = tid >> 5;

  __shared__ float red_m[8], red_s[8];
  __shared__ float sh_xb, sh_xl;

  const v4f ld = *reinterpret_cast<const v4f*>(LD + (size_t)bu * V_ + tid * 4);
  const int tgt = (u < U_) ? targets[b * U_ + u] : -1;

  for (int t = 0; t < T_; ++t) {
    const v4f le = *reinterpret_cast<const v4f*>(
        LE + (size_t)(b * T_ + t) * V_ + tid * 4);
    const float x0 = le[0] + ld[0], x1 = le[1] + ld[1];
    const float x2 = le[2] + ld[2], x3 = le[3] + ld[3];

    if (tid == (BLANK_ >> 2)) sh_xb = x3;       // BLANK=1023 -> tid 255, elem 3
    if (tgt >= 0 && (tgt >> 2) == tid) {
      const int e = tgt & 3;
      sh_xl = (e == 0) ? x0 : (e == 1) ? x1 : (e == 2) ? x2 : x3;
    }

    // lane-local online (max, sum-exp)
    float m = fmaxf(fmaxf(x0, x1), fmaxf(x2, x3));
    float s = __expf(x0 - m) + __expf(x1 - m) + __expf(x2 - m) + __expf(x3 - m);

    // wave32 reduction of (m, s) pairs
#pragma unroll
    for (int off = 16; off > 0; off >>= 1) {
      const float mo = __shfl_xor(m, off, 32);
      const float so = __shfl_xor(s, off, 32);
      const float mm = fmaxf(m, mo);
      s = s * __expf(m - mm) + so * __expf(mo - mm);
      m = mm;
    }
    if (lane == 0) { red_m[wave] = m; red_s[wave] = s; }
    __syncthreads();

    if (tid == 0) {
      float M2 = red_m[0], S2 = red_s[0];
#pragma unroll
      for (int w = 1; w < 8; ++w) {
        const float mo = red_m[w], so = red_s[w];
        const float mm = fmaxf(M2, mo);
        S2 = S2 * __expf(M2 - mm) + so * __expf(mo - mm);
        M2 = mm;
      }
      const float lse = M2 + __logf(S2);
      lpb[(size_t)(b * T_ + t) * (U_ + 1) + u] = sh_xb - lse;
      if (u < U_) lpl[(size_t)(b * T_ + t) * U_ + u] = sh_xl - lse;
    }
    __syncthreads();  // protect sh_* / red_* before next t
  }
}

// ---------------------------------------------------------------------------
// Kernel 3: RNN-T forward DP per batch, anti-diagonal wavefront, full alpha
// table resident in LDS (256*49*4 = 50176 B << 320 KB/WGP).
//   alpha[t][u] = logaddexp(alpha[t-1][u] + lpb[t-1][u],
//                           alpha[t][u-1] + lpl[t][u-1])
// ---------------------------------------------------------------------------
__global__ __launch_bounds__(64)
void rnnt_dp(const float* __restrict__ lpb, const float* __restrict__ lpl,
             const int* __restrict__ tlen, const int* __restrict__ ulen,
             float* __restrict__ loss) {
  const int b = blockIdx.x;
  const int u = threadIdx.x;  // lanes 0..U_ active
  __shared__ float alpha[T_ * (U_ + 1)];

  const float* pb = lpb + (size_t)b * T_ * (U_ + 1);
  const float* pl = lpl + (size_t)b * T_ * U_;

  for (int d = 0; d <= T_ - 1 + U_; ++d) {
    const int t = d - u;
    if (u <= U_ && t >= 0 && t < T_) {
      float val;
      if (t == 0) {
        val = (u == 0) ? 0.0f : alpha[u - 1] + pl[u - 1];
      } else if (u == 0) {
        val = alpha[(t - 1) * (U_ + 1)] + pb[(t - 1) * (U_ + 1)];
      } else {
        const float atop = alpha[(t - 1) * (U_ + 1) + u] + pb[(t - 1) * (U_ + 1) + u];
        const float alft = alpha[t * (U_ + 1) + (u - 1)] + pl[t * U_ + (u - 1)];
        const float mm = fmaxf(atop, alft);
        val = mm + log1pf(__expf(fminf(atop, alft) - mm));
      }
      alpha[t * (U_ + 1) + u] = val;
    }
    __syncthreads();
  }

  if (u == 0) {
    const int tl = tlen[b], ul = ulen[b];
    loss[b] = -(alpha[(tl - 1) * (U_ + 1) + ul] + pb[(tl - 1) * (U_ + 1) + ul]);
  }
}

// ---------------------------------------------------------------------------
// Launcher. Workspace layout (floats): LE | LD | lp_blank | lp_label  (~5.2 MB)
// ---------------------------------------------------------------------------
extern "C" void kernel_launch(void* const* d_in, const int* in_sizes, int n_in,
                              void* d_out, int out_size, void* d_ws, size_t ws_size,
                              hipStream_t stream) {
  const float* enc     = (const float*)d_in[0];  // (B, T, H)
  const float* dec     = (const float*)d_in[1];  // (B, U+1, H)
  const float* W       = (const float*)d_in[2];  // (V, H)
  const float* bias    = (const float*)d_in[3];  // (V,)
  const int*   targets = (const int*)d_in[4];    // (B, U)
  const int*   tl      = (const int*)d_in[5];    // (B,)
  const int*   ul      = (const int*)d_in[6];    // (B,)
  float*       out     = (float*)d_out;          // (B,)

  float* ws  = (float*)d_ws;
  float* LE  = ws;                                   // B*T*V
  float* LD  = LE  + (size_t)B_ * T_ * V_;           // B*(U+1)*V  (bias folded in)
  float* lpb = LD  + (size_t)B_ * (U_ + 1) * V_;     // B*T*(U+1)
  float* lpl = lpb + (size_t)B_ * T_ * (U_ + 1);     // B*T*U

  const dim3 blk(32 * GEMM_WAVES);
  const int  ncols = 16 * GEMM_NT * GEMM_WAVES;      // 128 columns per block
  const dim3 grdE((B_ * T_ + 15) / 16,        V_ / ncols);
  const dim3 grdD((B_ * (U_ + 1) + 15) / 16,  V_ / ncols);

  rnnt_wmma_gemm_nt<<<grdE, blk, 0, stream>>>(enc, W, nullptr, LE, B_ * T_);
  rnnt_wmma_gemm_nt<<<grdD, blk, 0, stream>>>(dec, W, bias,    LD, B_ * (U_ + 1));
  rnnt_lse<<<dim3(B_ * (U_ + 1)), dim3(256), 0, stream>>>(LE, LD, targets, lpb, lpl);
  rnnt_dp<<<dim3(B_), dim3(64), 0, stream>>>(lpb, lpl, tl, ul, out);
}